// ChessSSEA_86019605004695
// MI455X (gfx1250) — compile-verified
//
#include <hip/hip_runtime.h>
#include <hip/hip_bf16.h>

typedef float v2f __attribute__((ext_vector_type(2)));
typedef float v4f __attribute__((ext_vector_type(4)));
typedef float v8f __attribute__((ext_vector_type(8)));

// ---------------------------------------------------------------------------
// Workspace layout (floats)
// ---------------------------------------------------------------------------
#define WS_PT      0        // 64x64  PT[d*64+e] = piece_emb[e*64+d]
#define WS_W1T     4096     // 64x32  w1T[e*32+f] = w1[f*64+e]
#define WS_AVEC    6144     // 64     A[e] = sum_d P[e,d]^2
#define WS_NP      6208     // 64     N[p] = sum_d Q[p,d]^2
#define WS_M       6272     // 64x64  M[p*64+e] = sum_d Q[p,d]P[e,d]
#define WS_APRIME  10368    // 32     a'[f] = sum_e A[e] w1[f,e]
#define WS_MPRIME  10400    // 64x32  m'[p*32+f] = sum_e M[p,e] w1[f,e]
#define WS_NPRIME  12448    // 64x32  n'[p*32+f] = N[p]*sum_e w1[f,e] + b1[f]
#define WS_Y1      14496    // 4*128*64 relu(conv1(x))

// ---------------------------------------------------------------------------
// Kernel: prep — transpose piece_emb & w1, row self-dots of P and Q
// grid 160 x block 64
// ---------------------------------------------------------------------------
__global__ void k_prep(const float* __restrict__ P, const float* __restrict__ Q,
                       const float* __restrict__ w1,
                       float* __restrict__ PT, float* __restrict__ w1T,
                       float* __restrict__ Avec, float* __restrict__ Np)
{
    __shared__ float red[64];
    const int blk = blockIdx.x;
    const int t   = threadIdx.x;   // 0..63
    if (blk < 64) {                // PT transpose + Avec (row e = blk)
        float v = P[blk * 64 + t];
        PT[t * 64 + blk] = v;
        red[t] = v * v;
        __syncthreads();
        #pragma unroll
        for (int s = 32; s > 0; s >>= 1) {
            if (t < s) red[t] += red[t + s];
            __syncthreads();
        }
        if (t == 0) Avec[blk] = red[0];
    } else if (blk < 128) {        // Np (row p = blk-64)
        int p = blk - 64;
        float q = Q[p * 64 + t];
        red[t] = q * q;
        __syncthreads();
        #pragma unroll
        for (int s = 32; s > 0; s >>= 1) {
            if (t < s) red[t] += red[t + s];
            __syncthreads();
        }
        if (t == 0) Np[p] = red[0];
    } else {                       // w1 transpose (f = blk-128, e = t)
        int f = blk - 128;         // 0..31
        w1T[t * 32 + f] = w1[f * 64 + t];
    }
}

// ---------------------------------------------------------------------------
// Kernel: se writer — 268 MB streaming output, bandwidth bound.
// se[b,c,p,d,e] = x[b,c,p]*P[e,d] + Q[p,d]; one float4 of e per thread.
// grid 65536 x block 256
// ---------------------------------------------------------------------------
__global__ void k_se(const float* __restrict__ x, const float* __restrict__ PT,
                     const float* __restrict__ Q, float* __restrict__ se)
{
    long long t = (long long)blockIdx.x * 256 + threadIdx.x;  // 16,777,216 threads
    const int e4 = (int)(t & 15);
    long long r = t >> 4;
    const int d = (int)(r & 63); r >>= 6;
    const int p = (int)(r & 63); r >>= 6;
    const int c = (int)(r & 63);
    const int b = (int)(r >> 6);

    const float xv = x[((b * 65 + c) << 6) + p];
    const float q  = Q[(p << 6) + d];
    const v4f pc = *(const v4f*)(PT + (d << 6) + (e4 << 2));   // P[e4*4..+3, d]

    v4f o;
    o.x = fmaf(xv, pc.x, q);
    o.y = fmaf(xv, pc.y, q);
    o.z = fmaf(xv, pc.z, q);
    o.w = fmaf(xv, pc.w, q);
    __builtin_nontemporal_store(o, (v4f*)se + t);
}

// ---------------------------------------------------------------------------
// Kernel: f32 WMMA GEMM  C[M,N] = A[M,K] * B[K,N], K=64, one wave per 16x16
// tile of C.  v_wmma_f32_16x16x4_f32; ISA 16x4 f32 A layout:
//   lane<16: M=lane, K-pair (k0,k0+1);  lane>=16: M=lane-16, K-pair (k0+2,k0+3)
// B mirrored:  lane<16: N=lane, K-pair (k0,k0+1);  lane>=16: K-pair (+2,+3)
// grid (N/16, M/16) x block 32
// ---------------------------------------------------------------------------
__global__ void k_wmma_gemm64(const float* __restrict__ A, int lda,
                              const float* __restrict__ B, int ldb,
                              float* __restrict__ C, int ldc)
{
    const int lane = threadIdx.x;          // 0..31, EXEC all-1s
    const int half = lane >> 4;            // 0 or 1
    const int l16  = lane & 15;
    const int mt = blockIdx.y, nt = blockIdx.x;

    const float* Arow = A + (mt * 16 + l16) * lda;
    const float* Bcol = B + nt * 16 + l16;

    v8f acc = {};
    #pragma unroll
    for (int k0 = 0; k0 < 64; k0 += 4) {
        const int k = k0 + 2 * half;
        v2f a, b;
        a.x = Arow[k];
        a.y = Arow[k + 1];
        b.x = Bcol[k * ldb];
        b.y = Bcol[(k + 1) * ldb];
        acc = __builtin_amdgcn_wmma_f32_16x16x4_f32(
                  false, a, false, b, (short)0, acc, false, false);
    }
    #pragma unroll
    for (int i = 0; i < 8; ++i) {
        // VGPR i: lanes 0-15 -> M=i, lanes 16-31 -> M=i+8
        C[(mt * 16 + i + 8 * half) * ldc + nt * 16 + l16] = acc[i];
    }
}

// ---------------------------------------------------------------------------
// Kernel: fold — a'[f], n'[p,f]
// grid 1 x block 256
// ---------------------------------------------------------------------------
__global__ void k_fold(const float* __restrict__ Avec, const float* __restrict__ Np,
                       const float* __restrict__ w1, const float* __restrict__ b1,
                       float* __restrict__ aprime, float* __restrict__ nprime)
{
    __shared__ float s1[32];
    __shared__ float av[64];
    __shared__ float nps[64];
    const int t = threadIdx.x;
    if (t < 64)  av[t] = Avec[t];
    else if (t < 128) nps[t - 64] = Np[t - 64];
    __syncthreads();
    if (t < 32) {
        float sa = 0.f, ss = 0.f;
        for (int e = 0; e < 64; ++e) {
            float w = w1[t * 64 + e];
            sa = fmaf(av[e], w, sa);
            ss += w;
        }
        aprime[t] = sa;
        s1[t] = ss;
    }
    __syncthreads();
    for (int i = t; i < 2048; i += 256) {
        int p = i >> 5, f = i & 31;
        nprime[i] = fmaf(nps[p], s1[f], b1[f]);
    }
}

// ---------------------------------------------------------------------------
// Kernel: init rel_out with the 512*b2 constant (re-runs every launch)
// ---------------------------------------------------------------------------
__global__ void k_init_rel(float* __restrict__ rel, const float* __restrict__ b2)
{
    int i = blockIdx.x * 256 + threadIdx.x;
    if (i < 2048) rel[i] = 512.f * b2[0];
}

// ---------------------------------------------------------------------------
// Kernel: rel_out accumulation.  Block = one (b,p); thread c sums over s,f.
// enc(c,s) = sum_f w2[f]*relu(u*a'[f] + v*m'[p,f] + n'[p,f]); atomic over h.
// grid (64, 4) x block 64
// ---------------------------------------------------------------------------
__global__ void k_rel(const float* __restrict__ x, const float* __restrict__ aprime,
                      const float* __restrict__ mprime, const float* __restrict__ nprime,
                      const float* __restrict__ w2, float* __restrict__ rel_out)
{
    const int p = blockIdx.x;
    const int b = blockIdx.y;
    const int t = threadIdx.x;      // 0..63

    __shared__ float xv[64];
    __shared__ float ap[32], mp[32], np[32], wf[32];
    xv[t] = x[((b * 65 + t) << 6) + p];
    if (t < 32) {
        ap[t] = aprime[t];
        mp[t] = mprime[(p << 5) + t];
        np[t] = nprime[(p << 5) + t];
        wf[t] = w2[t];
    }
    __syncthreads();

    const float xc = xv[t];
    float acc = 0.f;
    for (int s = 0; s < 64; ++s) {
        const float xs = xv[s];
        const float u = xc * xs;
        const float v = xc + xs;
        #pragma unroll
        for (int f = 0; f < 32; ++f) {
            float h = fmaf(u, ap[f], fmaf(v, mp[f], np[f]));
            acc = fmaf(wf[f], fmaxf(h, 0.f), acc);
        }
    }
    atomicAdd(&rel_out[((b << 6) + t) << 3 | (p & 7)], acc);
}

// ---------------------------------------------------------------------------
// Kernels: the two 3x3 SAME convs (tiny: ~114 MFLOP total)
// y1 = relu(conv1(x));  ant = conv2(y1)
// ---------------------------------------------------------------------------
__global__ void k_conv1(const float* __restrict__ x, const float* __restrict__ w,
                        const float* __restrict__ bias, float* __restrict__ y)
{
    const int tid = blockIdx.x * 256 + threadIdx.x;  // 32768 outputs
    const int p = tid & 63;
    const int o = (tid >> 6) & 127;
    const int b = tid >> 13;
    const int h = p >> 3, ww = p & 7;
    float acc = bias[o];
    for (int c = 0; c < 65; ++c) {
        const float* xp = x + (b * 65 + c) * 64;
        const float* wp = w + (o * 65 + c) * 9;
        #pragma unroll
        for (int kh = 0; kh < 3; ++kh) {
            const int ih = h + kh - 1;
            if ((unsigned)ih < 8u) {
                #pragma unroll
                for (int kw = 0; kw < 3; ++kw) {
                    const int iw = ww + kw - 1;
                    if ((unsigned)iw < 8u)
                        acc = fmaf(xp[ih * 8 + iw], wp[kh * 3 + kw], acc);
                }
            }
        }
    }
    y[tid] = fmaxf(acc, 0.f);
}

__global__ void k_conv2(const float* __restrict__ y1, const float* __restrict__ w,
                        const float* __restrict__ bias, float* __restrict__ ant)
{
    const int tid = blockIdx.x * 256 + threadIdx.x;  // 32768 outputs
    const int p = tid & 63;
    const int o = (tid >> 6) & 127;
    const int b = tid >> 13;
    const int h = p >> 3, ww = p & 7;
    float acc = bias[o];
    for (int c = 0; c < 128; ++c) {
        const float* xp = y1 + (b * 128 + c) * 64;
        const float* wp = w + (o * 128 + c) * 9;
        #pragma unroll
        for (int kh = 0; kh < 3; ++kh) {
            const int ih = h + kh - 1;
            if ((unsigned)ih < 8u) {
                #pragma unroll
                for (int kw = 0; kw < 3; ++kw) {
                    const int iw = ww + kw - 1;
                    if ((unsigned)iw < 8u)
                        acc = fmaf(xp[ih * 8 + iw], wp[kh * 3 + kw], acc);
                }
            }
        }
    }
    ant[tid] = acc;
}

// ---------------------------------------------------------------------------
// Launch
// ---------------------------------------------------------------------------
extern "C" void kernel_launch(void* const* d_in, const int* in_sizes, int n_in,
                              void* d_out, int out_size, void* d_ws, size_t ws_size,
                              hipStream_t stream)
{
    const float* x    = (const float*)d_in[0];   // [4,65,8,8]
    const float* P    = (const float*)d_in[1];   // [64,64]  piece_emb[e,d]
    const float* Q    = (const float*)d_in[2];   // [64,64]  pos_emb[p,d]
    const float* w1   = (const float*)d_in[3];   // [32,64]
    const float* b1   = (const float*)d_in[4];   // [32]
    const float* w2   = (const float*)d_in[5];   // [1,32]
    const float* b2   = (const float*)d_in[6];   // [1]
    const float* c1w  = (const float*)d_in[7];   // [128,65,3,3]
    const float* c1b  = (const float*)d_in[8];   // [128]
    const float* c2w  = (const float*)d_in[9];   // [128,128,3,3]
    const float* c2b  = (const float*)d_in[10];  // [128]

    float* se_out  = (float*)d_out;              // 67,108,864
    float* rel_out = se_out + 67108864;          // 2,048
    float* ant_out = rel_out + 2048;             // 32,768

    float* ws = (float*)d_ws;
    float* PT     = ws + WS_PT;
    float* w1T    = ws + WS_W1T;
    float* Avec   = ws + WS_AVEC;
    float* Np     = ws + WS_NP;
    float* Mmat   = ws + WS_M;
    float* aprime = ws + WS_APRIME;
    float* mprime = ws + WS_MPRIME;
    float* nprime = ws + WS_NPRIME;
    float* y1     = ws + WS_Y1;

    // 1) prep: PT, w1T, Avec, Np
    k_prep<<<160, 64, 0, stream>>>(P, Q, w1, PT, w1T, Avec, Np);

    // 2) se (dominant 268MB streaming write; needs PT)
    k_se<<<65536, 256, 0, stream>>>(x, PT, Q, se_out);

    // 3) M = Q * P^T  (64x64x64, WMMA f32)
    k_wmma_gemm64<<<dim3(4, 4), 32, 0, stream>>>(Q, 64, PT, 64, Mmat, 64);

    // 4) m' = M * w1^T (64x32x64, WMMA f32)
    k_wmma_gemm64<<<dim3(2, 4), 32, 0, stream>>>(Mmat, 64, w1T, 32, mprime, 32);

    // 5) a', n'
    k_fold<<<1, 256, 0, stream>>>(Avec, Np, w1, b1, aprime, nprime);

    // 6) rel_out = 512*b2, then accumulate collapsed relationship module
    k_init_rel<<<8, 256, 0, stream>>>(rel_out, b2);
    k_rel<<<dim3(64, 4), 64, 0, stream>>>(x, aprime, mprime, nprime, w2, rel_out);

    // 7) anticipation convs
    k_conv1<<<128, 256, 0, stream>>>(x, c1w, c1b, y1);
    k_conv2<<<128, 256, 0, stream>>>(y1, c2w, c2b, ant_out);
}